// GCN_78726750535697
// MI455X (gfx1250) — compile-verified
//
#include <hip/hip_runtime.h>
#include <hip/hip_bf16.h>
#include <math.h>

#define N_NODES    50000
#define N_EDGES    800000
#define NFEAT      128
#define NHID       128
#define NCLASS     10
#define NUM_GRAPHS 512

typedef __attribute__((ext_vector_type(2))) float v2f;
typedef __attribute__((ext_vector_type(8))) float v8f;

// ---------------------------------------------------------------- utilities
__global__ void fill_kernel(float* __restrict__ p, float v, int n) {
    int i = blockIdx.x * blockDim.x + threadIdx.x;
    int stride = gridDim.x * blockDim.x;
    for (; i < n; i += stride) p[i] = v;
}

// deg[dst] += 1 per edge (deg pre-filled with 1.0 for the self loop)
__global__ void deg_kernel(const int* __restrict__ dst, float* __restrict__ deg, int ne) {
    int i = blockIdx.x * blockDim.x + threadIdx.x;
    if (i < ne) atomicAdd(&deg[dst[i]], 1.0f);
}

// ------------------------------------------------- fp32 WMMA GEMM  H = X @ W
// X: [nrows x 128], W: [128 x 128], H: [nrows x 128], nrows % 16 == 0.
// Block = 256 threads (8 waves). Block handles a 16-row stripe; wave w owns
// the 16-column tile starting at w*16. K-loop: 32 x V_WMMA_F32_16X16X4_F32.
__global__ __launch_bounds__(256) void gemm128_wmma(const float* __restrict__ X,
                                                    const float* __restrict__ W,
                                                    float* __restrict__ H) {
    const int wave    = threadIdx.x >> 5;
    const int lane    = threadIdx.x & 31;
    const int rowBase = blockIdx.x << 4;
    const int colBase = wave << 4;
    const int m  = lane & 15;          // row (A) / col (B,C)
    const int kh = (lane >> 4) << 1;   // lanes 16-31 hold K=2,3

    v8f acc = {};
    const float* ap = X + (size_t)(rowBase + m) * NFEAT + kh;
    const float* bp = W + (size_t)kh * NHID + colBase + m;

    for (int k0 = 0; k0 < NFEAT; k0 += 4) {
        v2f a, b;
        a.x = ap[k0];
        a.y = ap[k0 + 1];
        b.x = bp[(size_t)k0 * NHID];
        b.y = bp[(size_t)(k0 + 1) * NHID];
        acc = __builtin_amdgcn_wmma_f32_16x16x4_f32(
            /*neg_a=*/false, a, /*neg_b=*/false, b,
            /*c_mod=*/(short)0, acc, /*reuse_a=*/false, /*reuse_b=*/false);
    }

    // C/D layout: VGPR r -> row r (lanes 0-15) or row r+8 (lanes 16-31)
    const int rowOff = (lane >> 4) << 3;
    float* out = H + (size_t)(rowBase + rowOff) * NHID + colBase + m;
    #pragma unroll
    for (int r = 0; r < 8; ++r)
        out[(size_t)r * NHID] = acc[r];
}

// ------------------------------------------ edge scatter: msg[dst] += h[src]
// 32 lanes per edge; each lane moves 4 features (float4 gather + 4 f32 atomics
// that land in L2 — the 25.6 MB msg buffer is L2-resident on MI455X).
__global__ void scatter_kernel(const float* __restrict__ H, float* __restrict__ MSG,
                               const int* __restrict__ src, const int* __restrict__ dst,
                               int ne) {
    int t    = blockIdx.x * blockDim.x + threadIdx.x;
    int edge = t >> 5;
    int lane = t & 31;
    if (edge >= ne) return;
    int s = src[edge];
    int d = dst[edge];
    float4 v = ((const float4*)(H + (size_t)s * NHID))[lane];
    float* mp = MSG + (size_t)d * NHID + lane * 4;
    atomicAdd(mp + 0, v.x);
    atomicAdd(mp + 1, v.y);
    atomicAdd(mp + 2, v.z);
    atomicAdd(mp + 3, v.w);
}

// ---------------------------------------- x = (h + msg) / deg  (+ optional ReLU)
__global__ void combine_kernel(const float* __restrict__ H, const float* __restrict__ MSG,
                               const float* __restrict__ deg, float* __restrict__ OUT,
                               int doRelu) {
    int i = blockIdx.x * blockDim.x + threadIdx.x;
    if (i >= N_NODES * NHID) return;
    int row = i >> 7;  // NHID == 128
    float v = (H[i] + MSG[i]) / deg[row];
    if (doRelu) v = fmaxf(v, 0.0f);
    OUT[i] = v;
}

// ---------------------------- per-graph sum readout: gsum[g] += x[i], counts
__global__ void readout_kernel(const float* __restrict__ X, const int* __restrict__ gid,
                               float* __restrict__ gsum, float* __restrict__ counts) {
    int t    = blockIdx.x * blockDim.x + threadIdx.x;
    int node = t >> 5;
    int lane = t & 31;
    if (node >= N_NODES) return;
    int g = gid[node];
    float4 v = ((const float4*)(X + (size_t)node * NHID))[lane];
    float* gp = gsum + (size_t)g * NHID + lane * 4;
    atomicAdd(gp + 0, v.x);
    atomicAdd(gp + 1, v.y);
    atomicAdd(gp + 2, v.z);
    atomicAdd(gp + 3, v.w);
    if (lane == 0) atomicAdd(&counts[g], 1.0f);
}

// --------------- head: logits = gmean @ mlp_w + b ; log_softmax. 1 wave/graph
__global__ void head_kernel(const float* __restrict__ gsum, const float* __restrict__ counts,
                            const float* __restrict__ mlp_w, const float* __restrict__ mlp_b,
                            float* __restrict__ out) {
    int g    = blockIdx.x;
    int lane = threadIdx.x;  // 0..31, lanes 0..9 own a class
    float cnt = fmaxf(counts[g], 1.0f);
    float logit = -INFINITY;
    if (lane < NCLASS) {
        float acc = mlp_b[lane];
        const float* gp = gsum + (size_t)g * NHID;
        for (int k = 0; k < NHID; ++k)
            acc += (gp[k] / cnt) * mlp_w[k * NCLASS + lane];
        logit = acc;
    }
    // wave32 max reduction
    float mx = logit;
    for (int off = 16; off; off >>= 1)
        mx = fmaxf(mx, __shfl_xor(mx, off, 32));
    float e = (lane < NCLASS) ? expf(logit - mx) : 0.0f;
    for (int off = 16; off; off >>= 1)
        e += __shfl_xor(e, off, 32);
    float lse = logf(e);
    if (lane < NCLASS)
        out[(size_t)g * NCLASS + lane] = (logit - mx) - lse;
}

// ---------------------------------------------------------------------------
static inline size_t alignUp(size_t x, size_t a) { return (x + a - 1) & ~(a - 1); }

extern "C" void kernel_launch(void* const* d_in, const int* in_sizes, int n_in,
                              void* d_out, int out_size, void* d_ws, size_t ws_size,
                              hipStream_t stream) {
    const float* node_feat = (const float*)d_in[0];   // [N_NODES, NFEAT]
    const float* W1        = (const float*)d_in[1];   // [NFEAT, NHID]
    const float* W2        = (const float*)d_in[2];   // [NHID, NHID]
    const float* mlp_w     = (const float*)d_in[3];   // [NHID, NCLASS]
    const float* mlp_b     = (const float*)d_in[4];   // [NCLASS]
    const int*   edge_src  = (const int*)d_in[5];     // [N_EDGES]
    const int*   edge_dst  = (const int*)d_in[6];     // [N_EDGES]
    const int*   graph_ids = (const int*)d_in[7];     // [N_NODES]
    float* out = (float*)d_out;                       // [NUM_GRAPHS, NCLASS]

    // workspace carve-up
    const size_t featBytes = (size_t)N_NODES * NHID * sizeof(float);  // 25.6 MB
    char* ws = (char*)d_ws;
    size_t off = 0;
    float* H    = (float*)(ws + off); off = alignUp(off + featBytes, 256);
    float* MSG  = (float*)(ws + off); off = alignUp(off + featBytes, 256);
    float* X1   = (float*)(ws + off); off = alignUp(off + featBytes, 256);
    float* deg  = (float*)(ws + off); off = alignUp(off + (size_t)N_NODES * sizeof(float), 256);
    float* gsum = (float*)(ws + off); off = alignUp(off + (size_t)NUM_GRAPHS * NHID * sizeof(float), 256);
    float* cnts = (float*)(ws + off); off = alignUp(off + (size_t)NUM_GRAPHS * sizeof(float), 256);
    (void)ws_size; (void)n_in; (void)in_sizes; (void)out_size;

    const int TPB = 256;
    const int nElem      = N_NODES * NHID;                 // 6,400,000
    const int elemBlocks = (nElem + TPB - 1) / TPB;        // 25,000
    const int edgeBlocks = (N_EDGES + TPB - 1) / TPB;      // 3,125
    const int scatBlocks = (N_EDGES * 32) / TPB;           // 100,000
    const int rdBlocks   = (N_NODES * 32 + TPB - 1) / TPB; // 6,250
    const int gemmBlocks = N_NODES / 16;                   // 3,125 (exact)

    // degrees: deg = 1 + in-degree
    fill_kernel<<<(N_NODES + TPB - 1) / TPB, TPB, 0, stream>>>(deg, 1.0f, N_NODES);
    deg_kernel<<<edgeBlocks, TPB, 0, stream>>>(edge_dst, deg, N_EDGES);

    // ---- layer 1: x1 = relu((X@W1 + scatter) / deg)
    gemm128_wmma<<<gemmBlocks, TPB, 0, stream>>>(node_feat, W1, H);
    fill_kernel<<<elemBlocks, TPB, 0, stream>>>(MSG, 0.0f, nElem);
    scatter_kernel<<<scatBlocks, TPB, 0, stream>>>(H, MSG, edge_src, edge_dst, N_EDGES);
    combine_kernel<<<elemBlocks, TPB, 0, stream>>>(H, MSG, deg, X1, /*relu=*/1);

    // ---- layer 2: x2 = (X1@W2 + scatter) / deg   (combine in-place into H)
    gemm128_wmma<<<gemmBlocks, TPB, 0, stream>>>(X1, W2, H);
    fill_kernel<<<elemBlocks, TPB, 0, stream>>>(MSG, 0.0f, nElem);
    scatter_kernel<<<scatBlocks, TPB, 0, stream>>>(H, MSG, edge_src, edge_dst, N_EDGES);
    combine_kernel<<<elemBlocks, TPB, 0, stream>>>(H, MSG, deg, H, /*relu=*/0);

    // ---- per-graph mean readout + head
    fill_kernel<<<(NUM_GRAPHS * NHID + NUM_GRAPHS + TPB - 1) / TPB, TPB, 0, stream>>>(
        gsum, 0.0f, NUM_GRAPHS * NHID + NUM_GRAPHS);  // gsum and cnts are contiguous
    readout_kernel<<<rdBlocks, TPB, 0, stream>>>(H, graph_ids, gsum, cnts);
    head_kernel<<<NUM_GRAPHS, 32, 0, stream>>>(gsum, cnts, mlp_w, mlp_b, out);
}